// GNN_13657996001656
// MI455X (gfx1250) — compile-verified
//
#include <hip/hip_runtime.h>

#define H_DIM 128
#define GK 128          // K dimension of all GEMMs in this net
#define LDS_STRIDE 132  // padded row stride in floats: 132 % 64 = 4 -> bank-conflict-free

typedef __attribute__((ext_vector_type(2))) float v2f;
typedef __attribute__((ext_vector_type(8))) float v8f;

#if __has_builtin(__builtin_amdgcn_global_load_async_to_lds_b128)
#define GNN_ASYNC_STAGE 1
typedef int gnn_v4i __attribute__((ext_vector_type(4)));
typedef __attribute__((address_space(1))) gnn_v4i* gnn_as1_v4i;  // global v4i*
typedef __attribute__((address_space(3))) gnn_v4i* gnn_as3_v4i;  // LDS v4i*
#endif

__device__ __forceinline__ void atomAddF(float* p, float v) {
  __hip_atomic_fetch_add(p, v, __ATOMIC_RELAXED, __HIP_MEMORY_SCOPE_AGENT);
}

// ---------------- small utility kernels ----------------

__global__ void __launch_bounds__(256) gnn_fill_f32(float* p, float v, size_t n) {
  size_t i = (size_t)blockIdx.x * blockDim.x + threadIdx.x;
  size_t stride = (size_t)gridDim.x * blockDim.x;
  for (; i < n; i += stride) p[i] = v;
}

__global__ void __launch_bounds__(256) gnn_fill_i32(int* p, int v, size_t n) {
  size_t i = (size_t)blockIdx.x * blockDim.x + threadIdx.x;
  size_t stride = (size_t)gridDim.x * blockDim.x;
  for (; i < n; i += stride) p[i] = v;
}

__global__ void __launch_bounds__(256) gnn_count_deg(const int* __restrict__ dst,
                                                     int* __restrict__ degi, int E) {
  int e = blockIdx.x * 256 + threadIdx.x;
  if (e < E) atomicAdd(&degi[dst[e]], 1);
}

__global__ void __launch_bounds__(256) gnn_inv_deg(const int* __restrict__ degi,
                                                   float* __restrict__ invd, int N) {
  int i = blockIdx.x * 256 + threadIdx.x;
  if (i < N) { int d = degi[i]; invd[i] = d > 0 ? 1.0f / (float)d : 0.0f; }
}

// ---------------- exclusive scan of degrees -> CSR starts ----------------
// chunk = 1024 elements per 256-thread block

__global__ void __launch_bounds__(256) gnn_scan_partial(const int* __restrict__ degi,
                                                        int* __restrict__ bsum, int N) {
  __shared__ int sh[256];
  int base = blockIdx.x * 1024;
  int t = threadIdx.x;
  int s = 0;
#pragma unroll
  for (int j = 0; j < 4; ++j) {
    int idx = base + t * 4 + j;
    if (idx < N) s += degi[idx];
  }
  sh[t] = s;
  __syncthreads();
  for (int o = 128; o; o >>= 1) {
    if (t < o) sh[t] += sh[t + o];
    __syncthreads();
  }
  if (t == 0) bsum[blockIdx.x] = sh[0];
}

__global__ void gnn_scan_bsum(int* bsum, int nb) {
  if (threadIdx.x == 0 && blockIdx.x == 0) {
    int acc = 0;
    for (int i = 0; i < nb; ++i) { int v = bsum[i]; bsum[i] = acc; acc += v; }
  }
}

__global__ void __launch_bounds__(256) gnn_scan_final(const int* __restrict__ degi,
                                                      const int* __restrict__ bsum,
                                                      int* __restrict__ start,
                                                      int* __restrict__ cursor, int N) {
  __shared__ int sh[256];
  int base = blockIdx.x * 1024;
  int t = threadIdx.x;
  int loc[4];
  int s = 0;
#pragma unroll
  for (int j = 0; j < 4; ++j) {
    int idx = base + t * 4 + j;
    int v = (idx < N) ? degi[idx] : 0;
    loc[j] = s;
    s += v;
  }
  sh[t] = s;
  __syncthreads();
  for (int o = 1; o < 256; o <<= 1) {  // inclusive Hillis-Steele over thread sums
    int v = (t >= o) ? sh[t - o] : 0;
    __syncthreads();
    sh[t] += v;
    __syncthreads();
  }
  int thread_excl = sh[t] - s;
  int b = bsum[blockIdx.x];
#pragma unroll
  for (int j = 0; j < 4; ++j) {
    int idx = base + t * 4 + j;
    if (idx < N) {
      int v = b + thread_excl + loc[j];
      start[idx] = v;
      cursor[idx] = v;
    }
  }
}

__global__ void __launch_bounds__(256) gnn_fill_csr(const int* __restrict__ src,
                                                    const int* __restrict__ dst,
                                                    int* __restrict__ cursor,
                                                    int* __restrict__ csr, int E) {
  int e = blockIdx.x * 256 + threadIdx.x;
  if (e >= E) return;
  int pos = atomicAdd(&cursor[dst[e]], 1);
  csr[pos] = src[e];
}

// gather-aggregate: one wave per node, float4/lane, no fp atomics
__global__ void __launch_bounds__(256) gnn_aggregate(const float* __restrict__ z,
                                                     const int* __restrict__ csr,
                                                     const int* __restrict__ start,
                                                     const int* __restrict__ degi,
                                                     float* __restrict__ agg, int N) {
  int wv = threadIdx.x >> 5, lane = threadIdx.x & 31;
  int n = blockIdx.x * 8 + wv;
  if (n >= N) return;
  int s = start[n], c = degi[n];
  float4 acc = {0.0f, 0.0f, 0.0f, 0.0f};
  for (int j = 0; j < c; ++j) {
    int r = csr[s + j];  // wave-uniform
    float4 v = ((const float4*)(z + (size_t)r * H_DIM))[lane];
    acc.x += v.x; acc.y += v.y; acc.z += v.z; acc.w += v.w;
  }
  ((float4*)(agg + (size_t)n * H_DIM))[lane] = acc;
}

// l2-normalize rows of length 128 (codebook prep): one wave per row
__global__ void __launch_bounds__(256) gnn_l2norm_rows(const float* __restrict__ in,
                                                       float* __restrict__ out, int rows) {
  int wv = threadIdx.x >> 5, lane = threadIdx.x & 31;
  int r = blockIdx.x * 8 + wv;
  if (r >= rows) return;
  float4 v = ((const float4*)(in + (size_t)r * H_DIM))[lane];
  float s = v.x * v.x + v.y * v.y + v.z * v.z + v.w * v.w;
  for (int o = 16; o; o >>= 1) s += __shfl_xor(s, o, 32);
  float inv = 1.0f / (sqrtf(s) + 1e-8f);
  float4 ov = { v.x * inv, v.y * inv, v.z * inv, v.w * inv };
  ((float4*)(out + (size_t)r * H_DIM))[lane] = ov;
}

// z = relu(layernorm(h) * g + b), one wave per node
__global__ void __launch_bounds__(256) gnn_ln_relu(const float* __restrict__ in,
                                                   const float* __restrict__ g,
                                                   const float* __restrict__ b,
                                                   float* __restrict__ out, int Nn) {
  int wv = threadIdx.x >> 5, lane = threadIdx.x & 31;
  int n = blockIdx.x * 8 + wv;
  if (n >= Nn) return;
  float4 v = ((const float4*)(in + (size_t)n * H_DIM))[lane];
  float s = v.x + v.y + v.z + v.w;
  for (int o = 16; o; o >>= 1) s += __shfl_xor(s, o, 32);
  float mean = s * (1.0f / (float)H_DIM);
  float dx = v.x - mean, dy = v.y - mean, dz = v.z - mean, dw = v.w - mean;
  float vs = dx * dx + dy * dy + dz * dz + dw * dw;
  for (int o = 16; o; o >>= 1) vs += __shfl_xor(vs, o, 32);
  float inv = rsqrtf(vs * (1.0f / (float)H_DIM) + 1e-5f);
  float4 gg = ((const float4*)g)[lane];
  float4 bb = ((const float4*)b)[lane];
  float4 ov;
  ov.x = fmaxf(dx * inv * gg.x + bb.x, 0.0f);
  ov.y = fmaxf(dy * inv * gg.y + bb.y, 0.0f);
  ov.z = fmaxf(dz * inv * gg.z + bb.z, 0.0f);
  ov.w = fmaxf(dw * inv * gg.w + bb.w, 0.0f);
  ((float4*)(out + (size_t)n * H_DIM))[lane] = ov;
}

// ---------------- WMMA f32 GEMM with LDS-staged A strip ----------------
// out[M,Nout] (+)= (rs*A)[M,128] @ W[Nout,128]^T + bias
// 8 waves/block; the block's A rows are staged once into LDS (padded stride)
// via async global->LDS loads, then each wave runs V_WMMA_F32_16X16X4_F32.
__global__ void __launch_bounds__(256)
gnn_wmma_gemm(const float* __restrict__ A, const float* __restrict__ W,
              const float* __restrict__ bias, const float* __restrict__ row_scale,
              float* __restrict__ out, int M, int Nout, int accumulate) {
  __shared__ float As[32 * LDS_STRIDE];  // up to 32 staged rows (~16.9 KB)
  const int wv = threadIdx.x >> 5;
  const int lane = threadIdx.x & 31;
  const int tiles_n = Nout >> 4;        // 8 (H=128) or 4 (OUT=64)
  const int rows_pb = 128 / tiles_n;    // 16 or 32 rows staged per block
  const long tile0 = (long)blockIdx.x * 8;
  const int m0 = (int)(tile0 / tiles_n) * 16;
  int nrows = rows_pb;
  if (m0 + nrows > M) nrows = M - m0;

  // stage A[m0 .. m0+nrows) x 128 into LDS, 16B per thread-chunk
  const int nchunks = nrows * (GK / 4);
  for (int c = threadIdx.x; c < nchunks; c += 256) {
    int r = c >> 5;             // staged row
    int cc = (c & 31) * 4;      // column (floats)
#if GNN_ASYNC_STAGE
    __builtin_amdgcn_global_load_async_to_lds_b128(
        (gnn_as1_v4i)(A + (size_t)(m0 + r) * GK + cc),
        (gnn_as3_v4i)&As[r * LDS_STRIDE + cc], 0, 0);
#else
    *(float4*)&As[r * LDS_STRIDE + cc] =
        *(const float4*)(A + (size_t)(m0 + r) * GK + cc);
#endif
  }
#if GNN_ASYNC_STAGE
#if __has_builtin(__builtin_amdgcn_s_wait_asynccnt)
  __builtin_amdgcn_s_wait_asynccnt(0);
#else
  asm volatile("s_wait_asynccnt 0x0" ::: "memory");
#endif
#endif
  __syncthreads();

  const long tile = tile0 + wv;
  const int tile_m = (int)(tile / tiles_n);
  const int tile_n = (int)(tile % tiles_n);
  if (tile_m * 16 >= M) return;

  const int half = lane >> 4;
  const int lidx = lane & 15;
  const int row = tile_m * 16 + lidx;  // A row held by this lane
  const int col = tile_n * 16 + lidx;  // output column / W row held by this lane
  const float* arow = &As[(row - m0) * LDS_STRIDE];
  const float* wrow = W + (size_t)col * GK;
  const float rs = row_scale ? row_scale[row] : 1.0f;

  v8f acc = {};
  if (accumulate) {
#pragma unroll
    for (int j = 0; j < 8; ++j)
      acc[j] = out[(size_t)(tile_m * 16 + j + 8 * half) * Nout + col];
  }

  __builtin_prefetch(wrow, 0, 3);  // global_prefetch_b8: weights are L2-resident

#pragma unroll 4
  for (int k = 0; k < GK; k += 4) {
    int kk = k + 2 * half;
    float2 av = *(const float2*)(arow + kk);   // ds_load_b64
    float2 bv = *(const float2*)(wrow + kk);   // global_load_b64
    v2f a; a[0] = av.x * rs; a[1] = av.y * rs;
    v2f b; b[0] = bv.x;      b[1] = bv.y;
    acc = __builtin_amdgcn_wmma_f32_16x16x4_f32(
        false, a, false, b, (short)0, acc, false, false);
  }

  float bv = bias ? bias[col] : 0.0f;
#pragma unroll
  for (int j = 0; j < 8; ++j)
    out[(size_t)(tile_m * 16 + j + 8 * half) * Nout + col] = acc[j] + bv;
}

// ---------------- residual VQ: one wave per node, 3 stages, 16 codes ----------
// argmax(rn@cbn^T) == argmax(r@cbn^T): positive scale 1/(||r||+1e-8) is rank-preserving.
__global__ void __launch_bounds__(256)
gnn_vq(const float* __restrict__ h, const float* __restrict__ cbn /* [G,16,128] */,
       float* __restrict__ commit, float* __restrict__ ids, int layer, int Nn) {
  int wv = threadIdx.x >> 5, lane = threadIdx.x & 31;
  int n = blockIdx.x * 8 + wv;
  if (n >= Nn) return;
  float4 r = ((const float4*)(h + (size_t)n * H_DIM))[lane];
  float cpart = 0.0f;
#pragma unroll
  for (int q = 0; q < 3; ++q) {
    const float* cb = cbn + (size_t)q * 16 * H_DIM;
    int best = 0;
    float bests = -3.402823466e38f;
    for (int c = 0; c < 16; ++c) {
      float4 cv = ((const float4*)(cb + (size_t)c * H_DIM))[lane];
      float d = r.x * cv.x + r.y * cv.y + r.z * cv.z + r.w * cv.w;
      for (int o = 16; o; o >>= 1) d += __shfl_xor(d, o, 32);
      if (d > bests) { bests = d; best = c; }  // first-max tiebreak like argmax
    }
    float4 qv = ((const float4*)(cb + (size_t)best * H_DIM))[lane];
    float4 dd = { r.x - qv.x, r.y - qv.y, r.z - qv.z, r.w - qv.w };
    cpart += dd.x * dd.x + dd.y * dd.y + dd.z * dd.z + dd.w * dd.w;
    r = dd;
    if (lane == 0) ids[(size_t)n * 9 + layer * 3 + q] = (float)best;
  }
  for (int o = 16; o; o >>= 1) cpart += __shfl_xor(cpart, o, 32);
  if (lane == 0)
    atomAddF(commit, cpart * (0.25f / ((float)H_DIM * (float)Nn)));
}

// ---------------- host orchestration ----------------

extern "C" void kernel_launch(void* const* d_in, const int* in_sizes, int n_in,
                              void* d_out, int out_size, void* d_ws, size_t ws_size,
                              hipStream_t stream) {
  const int IN = 128, H = 128, OUT = 64, L = 3, G = 3, C = 16;
  const float* x         = (const float*)d_in[0];
  const int*   edge      = (const int*)  d_in[1];
  const float* lin1_w    = (const float*)d_in[2];
  const float* lin1_b    = (const float*)d_in[3];
  const float* norms_g   = (const float*)d_in[4];
  const float* norms_b   = (const float*)d_in[5];
  const float* convl_w   = (const float*)d_in[6];
  const float* convl_b   = (const float*)d_in[7];
  const float* convr_w   = (const float*)d_in[8];
  const float* codebooks = (const float*)d_in[9];
  const float* fnorm_g   = (const float*)d_in[10];
  const float* fnorm_b   = (const float*)d_in[11];
  const float* lin2_w    = (const float*)d_in[12];
  const float* lin2_b    = (const float*)d_in[13];

  const int N = in_sizes[0] / IN;
  const int E = in_sizes[1] / 2;
  const int* srcp = edge;
  const int* dstp = edge + E;
  const int nb = (N + 1023) / 1024;  // scan blocks

  // workspace carve-out (256B aligned)
  char* ws = (char*)d_ws;
  size_t off = 0;
  auto carve = [&](size_t bytes) -> void* {
    void* p = ws + off;
    off = (off + bytes + 255) & ~(size_t)255;
    return p;
  };
  int*   degi   = (int*)  carve((size_t)N * 4);
  float* invd   = (float*)carve((size_t)N * 4);
  int*   startb = (int*)  carve((size_t)N * 4);
  int*   cursor = (int*)  carve((size_t)N * 4);
  int*   csr    = (int*)  carve((size_t)E * 4);
  int*   bsum   = (int*)  carve((size_t)(nb + 1) * 4);
  float* h      = (float*)carve((size_t)N * H * 4);
  float* z      = (float*)carve((size_t)N * H * 4);
  float* agg    = (float*)carve((size_t)N * H * 4);
  float* cbn    = (float*)carve((size_t)L * G * C * H * 4);

  float* outp    = (float*)d_out;
  float* commitp = outp + (size_t)N * OUT;
  float* idsp    = commitp + 1;

  // --- CSR build: degrees, exclusive scan, bucket fill ---
  gnn_fill_i32<<<512, 256, 0, stream>>>(degi, 0, (size_t)N);
  gnn_count_deg<<<(E + 255) / 256, 256, 0, stream>>>(dstp, degi, E);
  gnn_inv_deg<<<(N + 255) / 256, 256, 0, stream>>>(degi, invd, N);
  gnn_scan_partial<<<nb, 256, 0, stream>>>(degi, bsum, N);
  gnn_scan_bsum<<<1, 32, 0, stream>>>(bsum, nb);
  gnn_scan_final<<<nb, 256, 0, stream>>>(degi, bsum, startb, cursor, N);
  gnn_fill_csr<<<(E + 255) / 256, 256, 0, stream>>>(srcp, dstp, cursor, csr, E);

  // pre-normalized codebooks; zero commit-loss accumulator
  gnn_l2norm_rows<<<(L * G * C + 7) / 8, 256, 0, stream>>>(codebooks, cbn, L * G * C);
  gnn_fill_f32<<<1, 256, 0, stream>>>(commitp, 0.0f, 1);

  // h = x @ lin1_w^T + lin1_b
  {
    long tiles = (long)(N / 16) * (H / 16);
    gnn_wmma_gemm<<<(unsigned)((tiles + 7) / 8), 256, 0, stream>>>(
        x, lin1_w, lin1_b, nullptr, h, N, H, 0);
  }

  for (int l = 0; l < L; ++l) {
    gnn_ln_relu<<<(N + 7) / 8, 256, 0, stream>>>(h, norms_g + l * H, norms_b + l * H, z, N);
    gnn_aggregate<<<(N + 7) / 8, 256, 0, stream>>>(z, csr, startb, degi, agg, N);

    long tiles = (long)(N / 16) * (H / 16);
    // h = (inv_deg * agg) @ convl_w[l]^T + convl_b[l]  (mean-aggr fused as row scale)
    gnn_wmma_gemm<<<(unsigned)((tiles + 7) / 8), 256, 0, stream>>>(
        agg, convl_w + (size_t)l * H * H, convl_b + l * H, invd, h, N, H, 0);
    // h += z @ convr_w[l]^T
    gnn_wmma_gemm<<<(unsigned)((tiles + 7) / 8), 256, 0, stream>>>(
        z, convr_w + (size_t)l * H * H, nullptr, nullptr, h, N, H, 1);

    gnn_vq<<<(N + 7) / 8, 256, 0, stream>>>(h, cbn + (size_t)l * G * C * H, commitp, idsp, l, N);
  }

  // out = relu(LN(h)) @ lin2_w^T + lin2_b
  gnn_ln_relu<<<(N + 7) / 8, 256, 0, stream>>>(h, fnorm_g, fnorm_b, z, N);
  {
    long tiles = (long)(N / 16) * (OUT / 16);
    gnn_wmma_gemm<<<(unsigned)((tiles + 7) / 8), 256, 0, stream>>>(
        z, lin2_w, lin2_b, nullptr, outp, N, OUT, 0);
  }
}